// GCNmodel_45595372814387
// MI455X (gfx1250) — compile-verified
//
#include <hip/hip_runtime.h>
#include <hip/hip_bf16.h>

// ---------------------------------------------------------------------------
// GCN (2x GCNConv + GraphNorm + mean-pool + MLP) for MI455X (gfx1250).
// Dense 100000x192 @ 192x192 GEMM runs on V_WMMA_F32_16X16X4_F32 (exact fp32).
// ---------------------------------------------------------------------------

#define N_NODES 100000
#define N_EDGES 1600000
#define N_GRAPHS 512
#define HID 192
#define HID2 96
#define EPS 1e-5f

typedef float v2f __attribute__((ext_vector_type(2)));
typedef float v8f __attribute__((ext_vector_type(8)));

// ---------------- degree / norm precompute ----------------

__global__ void k_init_deg(float* deg, int n) {
    int i = blockIdx.x * blockDim.x + threadIdx.x;
    if (i < n) deg[i] = 1.0f;                       // self-loop
}

__global__ void k_deg_edges(const int* __restrict__ col, float* deg, int e) {
    int i = blockIdx.x * blockDim.x + threadIdx.x;
    if (i < e) atomicAdd(deg + col[i], 1.0f);
}

__global__ void k_dinv(float* deg, int n) {
    int i = blockIdx.x * blockDim.x + threadIdx.x;
    if (i < n) deg[i] = rsqrtf(deg[i]);             // deg >= 1 always
}

__global__ void k_norm_e(const int* __restrict__ row, const int* __restrict__ col,
                         const float* __restrict__ dinv, float* __restrict__ nrm, int e) {
    int i = blockIdx.x * blockDim.x + threadIdx.x;
    if (i < e) nrm[i] = dinv[row[i]] * dinv[col[i]];
}

// ---------------- graph ranges (batch is sorted) ----------------

__global__ void k_zero_ranges(int* gs, int* ge, int g) {
    int i = blockIdx.x * blockDim.x + threadIdx.x;
    if (i < g) { gs[i] = 0; ge[i] = 0; }
}

__global__ void k_ranges(const int* __restrict__ batch, int* gs, int* ge, int n) {
    int i = blockIdx.x * blockDim.x + threadIdx.x;
    if (i >= n) return;
    int b = batch[i];
    if (i == 0     || batch[i - 1] != b) gs[b] = i;
    if (i == n - 1 || batch[i + 1] != b) ge[b] = i + 1;
}

// ---------------- conv1 (factorized: scalar aggregate) ----------------

__global__ void k_agg1_init(const float* __restrict__ x, const float* __restrict__ dinv,
                            float* __restrict__ agg, int n) {
    int i = blockIdx.x * blockDim.x + threadIdx.x;
    if (i < n) agg[i] = x[i] * dinv[i] * dinv[i];   // self-loop term
}

__global__ void k_agg1_edges(const int* __restrict__ row, const int* __restrict__ col,
                             const float* __restrict__ nrm, const float* __restrict__ x,
                             float* agg, int e) {
    int i = blockIdx.x * blockDim.x + threadIdx.x;
    if (i < e) atomicAdd(agg + col[i], x[row[i]] * nrm[i]);
}

// graph_norm(agg*W1+b1) + relu  ->  H[n,192]
__global__ __launch_bounds__(HID) void k_gn1(
    const float* __restrict__ agg, const int* __restrict__ gs, const int* __restrict__ ge,
    const float* __restrict__ W1, const float* __restrict__ b1,
    const float* __restrict__ g1, const float* __restrict__ be1, const float* __restrict__ a1,
    float* __restrict__ out) {
    int g = blockIdx.x, f = threadIdx.x;
    int s = gs[g], e = ge[g];
    float w = W1[f], bb = b1[f];
    float s1 = 0.f, s2 = 0.f;
    for (int n = s; n < e; n++) { float v = fmaf(agg[n], w, bb); s1 += v; s2 += v * v; }
    float cnt = (e - s) > 0 ? (float)(e - s) : 1.0f;
    float m = s1 / cnt;
    float al = a1[f];
    float var = s2 / cnt - m * m * (2.f * al - al * al);
    float rstd = rsqrtf(var + EPS);
    float gam = g1[f], bet = be1[f];
    for (int n = s; n < e; n++) {
        float v = fmaf(agg[n], w, bb);
        float o = fmaf(gam, (v - al * m) * rstd, bet);
        out[(size_t)n * HID + f] = fmaxf(o, 0.f);
    }
}

// ---------------- WMMA fp32 GEMM: C[M,192] = A[M,192] @ W[192,192] ----------------
// One wave per 16-row stripe x 4 N-tiles. K-steps of 4 with V_WMMA_F32_16X16X4_F32.

#define NTPW 4   // N-tiles per wave (12 total -> gridDim.y = 3)

__global__ __launch_bounds__(32) void k_gemm_wmma(
    const float* __restrict__ A, const float* __restrict__ W, float* __restrict__ C) {
    int tm  = blockIdx.x;                  // 0..6249
    int tn0 = blockIdx.y * NTPW;           // 0,4,8
    int lane = threadIdx.x;                // 0..31
    int mrow  = tm * 16 + (lane & 15);
    int khalf = (lane >> 4) * 2;           // A/B K sub-lane: 0 or 2
    int ncol  = lane & 15;

    v8f acc[NTPW];
#pragma unroll
    for (int t = 0; t < NTPW; t++) acc[t] = {};

    const float* Arow = A + (size_t)mrow * HID;
#pragma unroll 4
    for (int kk = 0; kk < HID; kk += 4) {
        v2f a;
        a.x = Arow[kk + khalf];
        a.y = Arow[kk + khalf + 1];
#pragma unroll
        for (int t = 0; t < NTPW; t++) {
            int n = (tn0 + t) * 16 + ncol;
            v2f b;
            b.x = W[(size_t)(kk + khalf)     * HID + n];
            b.y = W[(size_t)(kk + khalf + 1) * HID + n];
            acc[t] = __builtin_amdgcn_wmma_f32_16x16x4_f32(
                false, a, false, b, (short)0, acc[t], false, false);
        }
    }
    // C/D layout: VGPR v -> M = v + (lane>>4)*8, N = lane&15
    int mbase = tm * 16 + (lane >> 4) * 8;
#pragma unroll
    for (int t = 0; t < NTPW; t++) {
        int n = (tn0 + t) * 16 + ncol;
#pragma unroll
        for (int v = 0; v < 8; v++)
            C[(size_t)(mbase + v) * HID + n] = acc[t][v];
    }
}

// ---------------- conv2 aggregate ----------------

__global__ __launch_bounds__(HID) void k_conv2_init(
    const float* __restrict__ H, const float* __restrict__ dinv,
    const float* __restrict__ b2, float* __restrict__ out) {
    int i = blockIdx.x, f = threadIdx.x;
    float d = dinv[i];
    out[(size_t)i * HID + f] = fmaf(H[(size_t)i * HID + f], d * d, b2[f]);
}

#define EB 16
__global__ __launch_bounds__(HID) void k_conv2_edges(
    const int* __restrict__ row, const int* __restrict__ col,
    const float* __restrict__ nrm, const float* __restrict__ H,
    float* __restrict__ out, int e) {
    __shared__ int   sr[EB], sc[EB];
    __shared__ float sn[EB];
    int e0 = blockIdx.x * EB;
    int t = threadIdx.x;
    if (t < EB) {
        int ed = e0 + t;
        if (ed < e) { sr[t] = row[ed]; sc[t] = col[ed]; sn[t] = nrm[ed]; }
    }
    __syncthreads();
    int nm = e - e0; if (nm > EB) nm = EB;
    for (int j = 0; j < nm; j++) {
        float v = H[(size_t)sr[j] * HID + t] * sn[j];
        atomicAdd(out + (size_t)sc[j] * HID + t, v);
    }
}

// ---------------- graph_norm2 + relu + mean-pool (fused) ----------------

__global__ __launch_bounds__(HID) void k_gn2_pool(
    const float* __restrict__ H, const int* __restrict__ gs, const int* __restrict__ ge,
    const float* __restrict__ g2, const float* __restrict__ be2, const float* __restrict__ a2,
    float* __restrict__ pooled) {
    int g = blockIdx.x, f = threadIdx.x;
    int s = gs[g], e = ge[g];
    float s1 = 0.f, s2 = 0.f;
    for (int n = s; n < e; n++) { float v = H[(size_t)n * HID + f]; s1 += v; s2 += v * v; }
    float cnt = (e - s) > 0 ? (float)(e - s) : 1.0f;
    float m = s1 / cnt;
    float al = a2[f];
    float var = s2 / cnt - m * m * (2.f * al - al * al);
    float rstd = rsqrtf(var + EPS);
    float gam = g2[f], bet = be2[f];
    float ps = 0.f;
    for (int n = s; n < e; n++) {
        float v = H[(size_t)n * HID + f];
        float o = fmaf(gam, (v - al * m) * rstd, bet);
        ps += fmaxf(o, 0.f);
    }
    pooled[(size_t)g * HID + f] = ps / cnt;
}

// ---------------- tiny MLP head ----------------

__global__ __launch_bounds__(HID2) void k_mlp1(
    const float* __restrict__ pooled, const float* __restrict__ Wc1,
    const float* __restrict__ bc1, float* __restrict__ z) {
    int g = blockIdx.x, j = threadIdx.x;
    float acc = bc1[j];
    const float* p = pooled + (size_t)g * HID;
    for (int k = 0; k < HID; k++) acc = fmaf(p[k], Wc1[k * HID2 + j], acc);
    z[(size_t)g * HID2 + j] = fmaxf(acc, 0.f);
}

__global__ __launch_bounds__(32) void k_mlp2(
    const float* __restrict__ z, const float* __restrict__ Wc2,
    const float* __restrict__ bc2, float* __restrict__ out) {
    int g = blockIdx.x, j = threadIdx.x;
    if (j < 4) {
        float acc = bc2[j];
        const float* zz = z + (size_t)g * HID2;
        for (int k = 0; k < HID2; k++) acc = fmaf(zz[k], Wc2[k * 4 + j], acc);
        out[(size_t)g * 4 + j] = acc;
    }
}

// ---------------------------------------------------------------------------

extern "C" void kernel_launch(void* const* d_in, const int* in_sizes, int n_in,
                              void* d_out, int out_size, void* d_ws, size_t ws_size,
                              hipStream_t stream) {
    const float* x    = (const float*)d_in[0];
    const int*   eidx = (const int*)d_in[1];     // [2, E]
    const int*   batch= (const int*)d_in[2];
    const float* W1   = (const float*)d_in[3];
    const float* b1   = (const float*)d_in[4];
    const float* g1   = (const float*)d_in[5];
    const float* be1  = (const float*)d_in[6];
    const float* a1   = (const float*)d_in[7];
    const float* W2   = (const float*)d_in[8];
    const float* b2   = (const float*)d_in[9];
    const float* g2   = (const float*)d_in[10];
    const float* be2  = (const float*)d_in[11];
    const float* a2   = (const float*)d_in[12];
    const float* Wc1  = (const float*)d_in[13];
    const float* bc1  = (const float*)d_in[14];
    const float* Wc2  = (const float*)d_in[15];
    const float* bc2  = (const float*)d_in[16];
    float* out = (float*)d_out;

    const int* row = eidx;
    const int* col = eidx + N_EDGES;

    // workspace carve-up
    float* bufA   = (float*)d_ws;                      // N*HID   (h1norm, then conv2 accum)
    float* bufB   = bufA + (size_t)N_NODES * HID;      // N*HID   (h2pre)
    float* dinv   = bufB + (size_t)N_NODES * HID;      // N       (deg -> dinv)
    float* agg    = dinv + N_NODES;                    // N
    float* nrm    = agg + N_NODES;                     // E
    int*   gs     = (int*)(nrm + N_EDGES);             // G
    int*   ge     = gs + N_GRAPHS;                     // G
    float* pooled = (float*)(ge + N_GRAPHS);           // G*HID
    float* zbuf   = pooled + (size_t)N_GRAPHS * HID;   // G*HID2

    const int T = 256;
    dim3 nb((N_NODES + T - 1) / T), ebk((N_EDGES + T - 1) / T);

    // degrees & symmetric norm
    k_init_deg<<<nb, T, 0, stream>>>(dinv, N_NODES);
    k_deg_edges<<<ebk, T, 0, stream>>>(col, dinv, N_EDGES);
    k_dinv<<<nb, T, 0, stream>>>(dinv, N_NODES);
    k_norm_e<<<ebk, T, 0, stream>>>(row, col, dinv, nrm, N_EDGES);

    // graph ranges
    k_zero_ranges<<<(N_GRAPHS + T - 1) / T, T, 0, stream>>>(gs, ge, N_GRAPHS);
    k_ranges<<<nb, T, 0, stream>>>(batch, gs, ge, N_NODES);

    // conv1 (factorized scalar aggregate) + graph_norm1 + relu -> bufA
    k_agg1_init<<<nb, T, 0, stream>>>(x, dinv, agg, N_NODES);
    k_agg1_edges<<<ebk, T, 0, stream>>>(row, col, nrm, x, agg, N_EDGES);
    k_gn1<<<N_GRAPHS, HID, 0, stream>>>(agg, gs, ge, W1, b1, g1, be1, a1, bufA);

    // h2pre = bufA @ W2 via fp32 WMMA -> bufB
    dim3 gg(N_NODES / 16, HID / (16 * NTPW));
    k_gemm_wmma<<<gg, 32, 0, stream>>>(bufA, W2, bufB);

    // conv2 aggregate: self-loop + bias init, then edge scatter -> bufA
    k_conv2_init<<<N_NODES, HID, 0, stream>>>(bufB, dinv, b2, bufA);
    k_conv2_edges<<<(N_EDGES + EB - 1) / EB, HID, 0, stream>>>(row, col, nrm, bufB, bufA, N_EDGES);

    // graph_norm2 + relu + mean pool -> pooled
    k_gn2_pool<<<N_GRAPHS, HID, 0, stream>>>(bufA, gs, ge, g2, be2, a2, pooled);

    // MLP head
    k_mlp1<<<N_GRAPHS, HID2, 0, stream>>>(pooled, Wc1, bc1, zbuf);
    k_mlp2<<<N_GRAPHS, 32, 0, stream>>>(zbuf, Wc2, bc2, out);
}